// v4m1E_ModifiedScailingComputation_11252814316017
// MI455X (gfx1250) — compile-verified
//
#include <hip/hip_runtime.h>
#include <hip/hip_bf16.h>
#include <math.h>

// Shapes: N=6, B=2, H=12, T=1024, D=768
// out[b,h,q,t] = (1/8) * sum_n qk[n,b,h,q,t] * exp(a[b,t,n])
// a = GELU(RMSNorm(emb) @ w1) @ w2 + bias

typedef float v2f __attribute__((ext_vector_type(2)));
typedef float v8f __attribute__((ext_vector_type(8)));
typedef float f4  __attribute__((ext_vector_type(4)));

#define D_DIM   768
#define T_DIM   1024
#define N_DIM   6
#define H_DIM   12
#define B_DIM   2
#define NPER    ((size_t)B_DIM * H_DIM * T_DIM * T_DIM)   // 25,165,824 elems per n-slice
#define HTT     (H_DIM * T_DIM * T_DIM)                   // 12,582,912 elems per b
#define EPS_F32 1.1920928955078125e-07f
#define SCALING 0.125f                                    // 1/sqrt(64)

// ---------------------------------------------------------------------------
// Kernel 1: per-(b,t) row MLP via V_WMMA_F32_16X16X4_F32.
// One wave (32 threads) per block; each wave owns a 16-row tile (M=16).
// B-matrix zero-padding (cols 6..15) done with value selects (v_cndmask),
// NOT conditional loads -> no EXEC divergence inside the WMMA loop.
// Writes scaled exp() results TRANSPOSED: es[(b*6 + n)*1024 + t].
// ---------------------------------------------------------------------------
__global__ __launch_bounds__(32) void mlp_es_wmma_kernel(
    const float* __restrict__ emb,   // [B*T, D]
    const float* __restrict__ g,     // [D]
    const float* __restrict__ w1,    // [D, N] row-major
    const float* __restrict__ w2,    // [N, N] row-major
    const float* __restrict__ bias,  // [N]
    float* __restrict__ es)          // [B, N, T] transposed weights
{
    __shared__ float s_ssq[32];
    __shared__ float s_rms[16];
    __shared__ float s_gelu[16 * 8];   // 16 rows x 6 cols, stride 8

    const int   lane  = threadIdx.x;              // 0..31
    const int   m     = lane & 15;                // A row / B column index
    const int   hi    = lane >> 4;                // lane half selects K pair
    const int   row0  = blockIdx.x * 16;          // global (b*T + t) row base
    const int   mc    = (m < N_DIM) ? m : (N_DIM - 1);   // clamped (safe) column
    const float cmask = (m < N_DIM) ? 1.0f : 0.0f;       // zero-pad mask

    const float* __restrict__ xrow = emb + (size_t)(row0 + m) * D_DIM;
    const float* __restrict__ w1c  = w1 + mc;     // column base, stride N_DIM
    const int kb = hi * 2;                        // per-lane K offset in a chunk

    v8f   acc = {0.f, 0.f, 0.f, 0.f, 0.f, 0.f, 0.f, 0.f};
    float ssq = 0.f;

    #pragma unroll 4
    for (int k0 = 0; k0 < D_DIM; k0 += 4) {
        const int k = k0 + kb;
        const float x0 = xrow[k];
        const float x1 = xrow[k + 1];
        ssq += x0 * x0 + x1 * x1;

        v2f a;                 // A frag: VGPR v -> K = k0 + 2*hi + v, M = lane&15
        a.x = x0 * g[k];
        a.y = x1 * g[k + 1];

        v2f b;                 // B frag: VGPR v -> K = k0 + 2*hi + v, N = lane&15
        b.x = cmask * w1c[(size_t)k * N_DIM];        // unconditional load + mask
        b.y = cmask * w1c[(size_t)(k + 1) * N_DIM];

        acc = __builtin_amdgcn_wmma_f32_16x16x4_f32(
            /*neg_a=*/false, a, /*neg_b=*/false, b,
            /*c_mod=*/(short)0, acc, /*reuse_a=*/false, /*reuse_b=*/false);
    }

    // Row sum-of-squares: lanes l and l+16 jointly cover a full row.
    s_ssq[lane] = ssq;
    __syncthreads();
    if (lane < 16) {
        const float tot = s_ssq[lane] + s_ssq[lane + 16];
        s_rms[lane] = rsqrtf(tot * (1.0f / (float)D_DIM) + EPS_F32);
    }
    __syncthreads();

    // Apply per-row rsqrt, exact GELU, stash [16 x 6] matrix to LDS.
    // C/D layout: acc[j] holds M = j + 8*hi, N = lane&15.
    #pragma unroll
    for (int j = 0; j < 8; ++j) {
        const int   mm = j + hi * 8;
        const float y  = acc[j] * s_rms[mm];
        const float ge = 0.5f * y * (1.0f + erff(y * 0.70710678118654752440f));
        if (m < N_DIM) s_gelu[mm * 8 + m] = ge;   // store-only divergence is fine
    }
    __syncthreads();

    // Second (tiny) GEMM + bias + exp + scaling. 96 outputs, 3 per lane.
    #pragma unroll
    for (int i = 0; i < 3; ++i) {
        const int idx = lane * 3 + i;        // 0..95
        const int mm  = idx / N_DIM;
        const int n2  = idx % N_DIM;
        float a2 = bias[n2];
        #pragma unroll
        for (int kk = 0; kk < N_DIM; ++kk)
            a2 += s_gelu[mm * 8 + kk] * w2[kk * N_DIM + n2];
        const float ev   = SCALING * expf(a2);
        const int   grow = row0 + mm;        // global row = b*T + t
        const int   bb   = grow >> 10;       // / 1024
        const int   tt   = grow & 1023;
        es[(size_t)(bb * N_DIM + n2) * T_DIM + tt] = ev;
    }
}

// ---------------------------------------------------------------------------
// Kernel 2: HBM-bound streaming weighted sum.
// Each thread: 8 consecutive t (2x float4), 6 NT qk loads + 6 cached weight
// loads per float4, NT stores. 604 MB read + 100 MB write -> ~30 us floor.
// ---------------------------------------------------------------------------
__global__ __launch_bounds__(256) void scale_sum_kernel(
    const float* __restrict__ qk,   // [N, B, H, T, T]
    const float* __restrict__ es,   // [B, N, T]
    float* __restrict__ out)        // [B, H, T, T]
{
    const unsigned gid  = blockIdx.x * 256u + threadIdx.x;
    const unsigned base = gid * 8u;                 // covers 8 consecutive t

    const int t  = (int)(base & (T_DIM - 1));       // 8-aligned within a row
    const int bb = (int)(base / (unsigned)HTT);     // batch index (0 or 1)
    const float* __restrict__ esb = es + (size_t)bb * (N_DIM * T_DIM) + t;

    f4 acc0 = {0.f, 0.f, 0.f, 0.f};
    f4 acc1 = {0.f, 0.f, 0.f, 0.f};

    #pragma unroll
    for (int n = 0; n < N_DIM; ++n) {
        const float* __restrict__ src = qk + (size_t)n * NPER;  // uniform base
        const f4 q0 = __builtin_nontemporal_load((const f4*)(src + base));
        const f4 q1 = __builtin_nontemporal_load((const f4*)(src + base + 4));
        const f4 w0 = *(const f4*)(esb + n * T_DIM);       // L2/WGP$ resident
        const f4 w1 = *(const f4*)(esb + n * T_DIM + 4);
        acc0 += q0 * w0;
        acc1 += q1 * w1;
    }

    __builtin_nontemporal_store(acc0, (f4*)(out + base));
    __builtin_nontemporal_store(acc1, (f4*)(out + base + 4));
}

// ---------------------------------------------------------------------------
extern "C" void kernel_launch(void* const* d_in, const int* in_sizes, int n_in,
                              void* d_out, int out_size, void* d_ws, size_t ws_size,
                              hipStream_t stream) {
    const float* qk   = (const float*)d_in[0];  // qk_stack [6,2,12,1024,1024]
    const float* emb  = (const float*)d_in[1];  // input_embedding [2,1024,768]
    const float* g    = (const float*)d_in[2];  // norm_weight [768]
    const float* w1   = (const float*)d_in[3];  // [768,6]
    const float* w2   = (const float*)d_in[4];  // [6,6]
    const float* bias = (const float*)d_in[5];  // [6]
    float* out = (float*)d_out;
    float* es  = (float*)d_ws;                  // needs B*N*T*4 = 48 KB

    // 2048 rows / 16 rows-per-wave = 128 single-wave blocks
    mlp_es_wmma_kernel<<<dim3(128), dim3(32), 0, stream>>>(emb, g, w1, w2, bias, es);

    // 25,165,824 outputs / 8 per thread / 256 per block = 12288 blocks (exact)
    scale_sum_kernel<<<dim3(12288), dim3(256), 0, stream>>>(qk, es, out);
}